// InvariantPointAttention_20194936225864
// MI455X (gfx1250) — compile-verified
//
#include <hip/hip_runtime.h>
#include <hip/hip_bf16.h>
#include <math.h>
#include <stdint.h>

#define LSEQ 768
#define DIN  384
#define DZ   128
#define NH   12
#define CD   16
#define NQP  4
#define NPP  6

#define C_SQU 0.25f
#define W_C   0.23570226039551584f   // sqrt(2/(9*4))

typedef float v2f __attribute__((ext_vector_type(2)));
typedef float v4f __attribute__((ext_vector_type(4)));
typedef float v8f __attribute__((ext_vector_type(8)));

// V_WMMA_F32_16X16X4_F32  (wave32): A 16x4 (2 VGPR), B 4x16 (2 VGPR), C/D 16x16 (8 VGPR)
static __device__ __forceinline__ v8f wmma_f32(v2f a, v2f b, v8f c) {
  return __builtin_amdgcn_wmma_f32_16x16x4_f32(false, a, false, b, (short)0, c, false, false);
}

// ---------------------------------------------------------------------------
// Row-major f32 GEMM: C [M,N] (+)= A [M,K] @ B [K,N].
// Requirements (all launches satisfy): M % 128 == 0, K % 8 == 0.
// Block tile: 128(M) x 32(N); wave w owns M-tile w, all 8 waves share one
// 32-wide B strip staged per 32-K chunk into LDS *transposed* so each B
// fragment is a single ds_load_b64. N edge handled by clamped (in-range)
// addresses + guarded stores; K edge (216) via a uniform partial-chunk path.
// ---------------------------------------------------------------------------
#define KC 32
__global__ __launch_bounds__(256) void ipa_gemm_f32(
    const float* __restrict__ A, const float* __restrict__ B,
    float* __restrict__ Cm, int M, int N, int K, int accumulate) {
  const int tid  = threadIdx.x;
  const int wave = tid >> 5;
  const int lane = tid & 31;
  const int half = lane >> 4;
  const int lr   = lane & 15;

  const int tiles_n = (N + 31) >> 5;
  const int bm = blockIdx.x / tiles_n;
  const int bn = blockIdx.x % tiles_n;
  const int m0 = bm * 128 + wave * 16;
  const int n0 = bn * 32;

  __shared__ float BsT[32 * 34];           // [c][k], padded (bank-conflict free)

  int  col[2], colc[2];
  bool cok[2];
  v8f  acc[2];
  for (int t = 0; t < 2; ++t) {
    col[t]  = n0 + t * 16 + lr;
    cok[t]  = col[t] < N;
    colc[t] = cok[t] ? col[t] : (N - 1);   // always in-range
    v8f zv = {0.f, 0.f, 0.f, 0.f, 0.f, 0.f, 0.f, 0.f};
    acc[t] = zv;
  }
  if (accumulate) {
    for (int t = 0; t < 2; ++t)
      for (int r = 0; r < 8; ++r)
        acc[t][r] = Cm[(m0 + r + 8 * half) * N + colc[t]];
  }

  const int ks = tid >> 5;                 // staging: k = ks + 8*it
  const int cs = tid & 31;                 // staging column
  const int ccs = (n0 + cs < N) ? (n0 + cs) : (N - 1);

  const float* Ap = A + (m0 + lr) * K + 2 * half;

  for (int k0 = 0; k0 < K; k0 += KC) {
    const int kch = (K - k0 < KC) ? (K - k0) : KC;
    __syncthreads();
#pragma unroll
    for (int it = 0; it < 4; ++it) {
      const int k = ks + it * 8;
      if (k < kch)                         // uniform across lanes of a "row"
        BsT[cs * 34 + k] = B[(size_t)(k0 + k) * N + ccs];
    }
    __syncthreads();
    if (kch == KC) {                       // fast path, fully unrolled
#pragma unroll
      for (int kk = 0; kk < KC; kk += 4) {
        const v2f a = *(const v2f*)(Ap + k0 + kk);           // global_load_b64
        const int ka = kk + 2 * half;
        for (int t = 0; t < 2; ++t) {
          const v2f b = *(const v2f*)&BsT[(t * 16 + lr) * 34 + ka]; // ds_load_b64
          acc[t] = wmma_f32(a, b, acc[t]);
        }
      }
    } else {                               // uniform K-edge path
      for (int kk = 0; kk < kch; kk += 4) {
        const v2f a = *(const v2f*)(Ap + k0 + kk);
        const int ka = kk + 2 * half;
        for (int t = 0; t < 2; ++t) {
          const v2f b = *(const v2f*)&BsT[(t * 16 + lr) * 34 + ka];
          acc[t] = wmma_f32(a, b, acc[t]);
        }
      }
    }
  }

  for (int t = 0; t < 2; ++t)
    if (cok[t])
      for (int r = 0; r < 8; ++r)
        Cm[(m0 + r + 8 * half) * N + col[t]] = acc[t][r];
}

// ---------------------------------------------------------------------------
// Point prep: rigid transforms + fold all non-z score terms into 32-wide
// vectors so score_pre[i,j,h] = dot32(Aext[i,h,:], Bext[j,h,:]).
// ---------------------------------------------------------------------------
__global__ void ipa_point_prep(
    const float* __restrict__ q1, const float* __restrict__ k1,
    const float* __restrict__ q2p, const float* __restrict__ k2p,
    const float* __restrict__ v3p, const float* __restrict__ rot,
    const float* __restrict__ trans, const float* __restrict__ gamma,
    float* __restrict__ Aext, float* __restrict__ Bext, float* __restrict__ v3g) {
  const int idx = blockIdx.x * blockDim.x + threadIdx.x;
  if (idx >= LSEQ * NH) return;
  const int l = idx / NH, h = idx % NH;
  const float g  = log1pf(expf(gamma[h]));   // softplus
  const float sc = g * W_C;
  const float* R = rot + (size_t)l * 9;
  const float* t = trans + (size_t)l * 3;
  float* Ae = Aext + (size_t)idx * 32;
  float* Be = Bext + (size_t)idx * 32;

  for (int c = 0; c < CD; ++c) {
    Ae[c] = C_SQU * q1[(size_t)l * (NH * CD) + h * CD + c];
    Be[c] =          k1[(size_t)l * (NH * CD) + h * CD + c];
  }
  float sq = 0.f, sk = 0.f;
  for (int q = 0; q < NQP; ++q) {
    const float* xq = q2p + (size_t)l * (NH * NQP * 3) + h * (NQP * 3) + q * 3;
    const float* xk = k2p + (size_t)l * (NH * NQP * 3) + h * (NQP * 3) + q * 3;
    for (int i = 0; i < 3; ++i) {
      const float gq = R[i*3+0]*xq[0] + R[i*3+1]*xq[1] + R[i*3+2]*xq[2] + t[i];
      const float gk = R[i*3+0]*xk[0] + R[i*3+1]*xk[1] + R[i*3+2]*xk[2] + t[i];
      sq += gq * gq; sk += gk * gk;
      Ae[16 + q*3 + i] = sc * gq;   // * k2 -> + g*W_C*cross
      Be[16 + q*3 + i] = gk;
    }
  }
  Ae[28] = -0.5f * sc * sq;  Be[28] = 1.f;
  Ae[29] = -0.5f * sc;       Be[29] = sk;
  Ae[30] = 0.f; Ae[31] = 0.f; Be[30] = 0.f; Be[31] = 0.f;

  for (int p = 0; p < NPP; ++p) {
    const float* xv = v3p + (size_t)l * (NH * NPP * 3) + h * (NPP * 3) + p * 3;
    for (int i = 0; i < 3; ++i)
      v3g[(size_t)l * (NH * NPP * 3) + h * (NPP * 3) + p * 3 + i] =
          R[i*3+0]*xv[0] + R[i*3+1]*xv[1] + R[i*3+2]*xv[2] + t[i];
  }
}

// ---------------------------------------------------------------------------
// Fused pass: one workgroup per query row i. Single streaming read of z via
// async global->LDS DMA (ASYNCcnt). Per 64-wide j tile:
//   z -> LDS (global_load_async_to_lds_b128);
//   bias = z@Wz (WMMA, waves 0-3) || raw score dots (waves 4-7);
//   att = score + bias;  o1 += att@z (WMMA, one d-tile per wave);
//   o2/o3 VALU accumulation from L2-resident v1/v3g.
// z padded ld=132, att ld=68 -> conflict-free A-fragment column reads.
// ---------------------------------------------------------------------------
__global__ __launch_bounds__(256) void ipa_fused(
    const float* __restrict__ z, const float* __restrict__ Wz,
    const float* __restrict__ Aext, const float* __restrict__ Bext,
    const float* __restrict__ v1, const float* __restrict__ v3g,
    const float* __restrict__ rot, const float* __restrict__ trans,
    float* __restrict__ o1w, float* __restrict__ o2w, float* __restrict__ o3w) {
  const int i    = blockIdx.x;
  const int tid  = threadIdx.x;
  const int wave = tid >> 5;
  const int lane = tid & 31;
  const int half = lane >> 4;
  const int lr   = lane & 15;

  __shared__ float zt[64 * 132];    // [j][d], padded
  __shared__ float WzL[128 * 16];   // [d][h], h padded to 16 with zeros
  __shared__ float biasT[64 * 16];  // [j][h]
  __shared__ float attL[16 * 68];   // [h][j], padded
  __shared__ float Arow[NH * 32];   // Aext row for this i

  for (int t = tid; t < 128 * 16; t += 256) {
    const int d = t >> 4, h = t & 15;
    WzL[t] = (h < NH) ? Wz[d * NH + h] : 0.f;
  }
  for (int t = tid; t < NH * 32; t += 256)
    Arow[t] = Aext[(size_t)i * (NH * 32) + t];
  for (int t = tid; t < 16 * 68; t += 256)
    attL[t] = 0.f;   // rows h=12..15 stay zero

  v8f acc1 = {0.f, 0.f, 0.f, 0.f, 0.f, 0.f, 0.f, 0.f};
  float acc2 = 0.f;
  float a3x = 0.f, a3y = 0.f, a3z = 0.f;
  const int d0 = wave * 16;
  const float* zr_base = z + (size_t)i * LSEQ * DZ;

  // this thread's fixed slots in the z tile copy (8 x 16B per thread)
  const int jl0 = tid >> 5;          // rows jl0 + 8*it
  const int c4  = (tid & 31) * 4;    // 16B column

  for (int j0 = 0; j0 < LSEQ; j0 += 64) {
    __syncthreads();                 // previous tile fully consumed
    {
      const float* zr = zr_base + (size_t)j0 * DZ;
#pragma unroll
      for (int it = 0; it < 8; ++it) {
        const int jl = jl0 + it * 8;
        const unsigned ldsa = (unsigned)(uintptr_t)&zt[jl * 132 + c4];
        const float*   ga   = zr + (size_t)jl * DZ + c4;
        asm volatile("global_load_async_to_lds_b128 %0, %1, off"
                     :: "v"(ldsa), "v"(ga) : "memory");
      }
      if (j0 + 64 < LSEQ)            // prefetch next tile into GL2
        __builtin_prefetch(zr_base + (size_t)(j0 + 64) * DZ + (size_t)tid * 32, 0, 0);
      asm volatile("s_wait_asynccnt 0" ::: "memory");
    }
    __syncthreads();

    if (wave < 4) {                  // bias_z tile [64 x 12] = z_tile @ Wz
      const int jt0 = wave * 16;
      const float* za = &zt[(jt0 + lr) * 132 + 2 * half];
      v8f bacc = {0.f, 0.f, 0.f, 0.f, 0.f, 0.f, 0.f, 0.f};
#pragma unroll 4
      for (int k0 = 0; k0 < DZ; k0 += 4) {
        const v2f a = *(const v2f*)(za + k0);                 // ds_load_b64
        const int ka = k0 + 2 * half;
        v2f b;
        b.x = WzL[ka * 16 + lr];
        b.y = WzL[(ka + 1) * 16 + lr];
        bacc = wmma_f32(a, b, bacc);
      }
      for (int r = 0; r < 8; ++r)
        biasT[(jt0 + r + 8 * half) * 16 + lr] = bacc[r];
    } else {                         // raw scores: dot32(Aext[i,h], Bext[j,h])
      for (int t = tid - 128; t < NH * 64; t += 128) {
        const int h = t >> 6, jl = t & 63;
        const v4f* Bv = (const v4f*)(Bext + ((size_t)(j0 + jl) * NH + h) * 32);
        const float* Av = Arow + h * 32;
        float acc = 0.f;
#pragma unroll
        for (int k4 = 0; k4 < 8; ++k4) {
          const v4f bb = Bv[k4];     // global_load_b128 (128B-aligned rows)
          acc += Av[k4*4+0]*bb.x + Av[k4*4+1]*bb.y
               + Av[k4*4+2]*bb.z + Av[k4*4+3]*bb.w;
        }
        attL[h * 68 + jl] = acc;
      }
    }
    __syncthreads();
    for (int t = tid; t < NH * 64; t += 256) {
      const int h = t >> 6, jl = t & 63;
      attL[h * 68 + jl] += biasT[jl * 16 + h];
    }
    __syncthreads();

    // o1: [12 x 128] += att[12 x 64] @ z[64 x 128]; wave owns d-tile d0
    {
      const float* aa = &attL[lr * 68 + 2 * half];
#pragma unroll 4
      for (int k0 = 0; k0 < 64; k0 += 4) {
        const v2f a = *(const v2f*)(aa + k0);                 // ds_load_b64
        const int ka = k0 + 2 * half;
        v2f b;
        b.x = zt[ka * 132 + d0 + lr];
        b.y = zt[(ka + 1) * 132 + d0 + lr];
        acc1 = wmma_f32(a, b, acc1);
      }
    }
    // o2[h,c] += sum_j att[h,j] * v1[j,h,c]   (v1 L2-resident)
    if (tid < NH * CD) {
      const int h = tid >> 4;
      const float* v1p  = v1 + (size_t)j0 * (NH * CD) + tid;
      const float* arow = attL + h * 68;
#pragma unroll 4
      for (int jl = 0; jl < 64; ++jl)
        acc2 += arow[jl] * v1p[(size_t)jl * (NH * CD)];
    }
    // o3[h,p,:] += sum_j att[h,j] * v3g[j,h,p,:]
    if (tid >= 184) {
      const int u = tid - 184;
      const int h = u / NPP, p = u % NPP;
      const float* vb   = v3g + (size_t)j0 * (NH * NPP * 3) + h * (NPP * 3) + p * 3;
      const float* arow = attL + h * 68;
#pragma unroll 4
      for (int jl = 0; jl < 64; ++jl) {
        const float a = arow[jl];
        const float* vv = vb + (size_t)jl * (NH * NPP * 3);
        a3x += a * vv[0]; a3y += a * vv[1]; a3z += a * vv[2];
      }
    }
  }

  for (int r = 0; r < 8; ++r) {      // o1 store [L, H*DZ]
    const int h = r + 8 * half;
    if (h < NH) o1w[(size_t)i * (NH * DZ) + h * DZ + d0 + lr] = acc1[r];
  }
  if (tid < NH * CD) o2w[(size_t)i * (NH * CD) + tid] = acc2;
  if (tid >= 184) {                  // inverse transform + store
    const int u = tid - 184;
    const int h = u / NPP, p = u % NPP;
    const float* R = rot + (size_t)i * 9;
    const float* t = trans + (size_t)i * 3;
    const float x0 = a3x - t[0], x1 = a3y - t[1], x2 = a3z - t[2];
    float* op = o3w + (size_t)i * (NH * NPP * 3) + h * (NPP * 3) + p * 3;
    op[0] = R[0]*x0 + R[3]*x1 + R[6]*x2;   // rot^T @ (x - t)
    op[1] = R[1]*x0 + R[4]*x1 + R[7]*x2;
    op[2] = R[2]*x0 + R[5]*x1 + R[8]*x2;
  }
}

__global__ void ipa_init_out(const float* __restrict__ b1, const float* __restrict__ b2,
                             const float* __restrict__ b3, float* __restrict__ out) {
  const int idx = blockIdx.x * blockDim.x + threadIdx.x;
  if (idx >= LSEQ * DIN) return;
  const int d = idx % DIN;
  out[idx] = b1[d] + b2[d] + b3[d];
}

// ---------------------------------------------------------------------------
extern "C" void kernel_launch(void* const* d_in, const int* in_sizes, int n_in,
                              void* d_out, int out_size, void* d_ws, size_t ws_size,
                              hipStream_t stream) {
  const float* s     = (const float*)d_in[0];
  const float* z     = (const float*)d_in[1];
  const float* rot   = (const float*)d_in[2];
  const float* trans = (const float*)d_in[3];
  const float* Wq1   = (const float*)d_in[4];
  const float* Wk1   = (const float*)d_in[5];
  const float* Wv1   = (const float*)d_in[6];
  const float* Wq2   = (const float*)d_in[7];
  const float* Wk2   = (const float*)d_in[8];
  const float* Wv3   = (const float*)d_in[9];
  const float* Wz    = (const float*)d_in[10];
  const float* W1    = (const float*)d_in[11];
  const float* b1    = (const float*)d_in[12];
  const float* W2    = (const float*)d_in[13];
  const float* b2    = (const float*)d_in[14];
  const float* W3    = (const float*)d_in[15];
  const float* b3    = (const float*)d_in[16];
  const float* gamma = (const float*)d_in[17];
  float* out = (float*)d_out;

  float* w = (float*)d_ws;
  float* q1  = w; w += LSEQ * NH * CD;
  float* k1  = w; w += LSEQ * NH * CD;
  float* v1  = w; w += LSEQ * NH * CD;
  float* q2p = w; w += LSEQ * NH * NQP * 3;
  float* k2p = w; w += LSEQ * NH * NQP * 3;
  float* v3p = w; w += LSEQ * NH * NPP * 3;
  float* Aex = w; w += LSEQ * NH * 32;
  float* Bex = w; w += LSEQ * NH * 32;
  float* v3g = w; w += LSEQ * NH * NPP * 3;
  float* o1w = w; w += LSEQ * NH * DZ;
  float* o2w = w; w += LSEQ * NH * CD;
  float* o3w = w; w += LSEQ * NH * NPP * 3;
  (void)in_sizes; (void)n_in; (void)out_size; (void)ws_size;

  auto gemm = [&](const float* A, const float* B, float* Cm,
                  int M, int N, int K, int accum) {
    const int blocks = (M / 128) * ((N + 31) / 32);
    ipa_gemm_f32<<<blocks, 256, 0, stream>>>(A, B, Cm, M, N, K, accum);
  };

  // Projections
  gemm(s, Wq1, q1,  LSEQ, NH * CD,      DIN, 0);
  gemm(s, Wk1, k1,  LSEQ, NH * CD,      DIN, 0);
  gemm(s, Wv1, v1,  LSEQ, NH * CD,      DIN, 0);
  gemm(s, Wq2, q2p, LSEQ, NH * NQP * 3, DIN, 0);
  gemm(s, Wk2, k2p, LSEQ, NH * NQP * 3, DIN, 0);
  gemm(s, Wv3, v3p, LSEQ, NH * NPP * 3, DIN, 0);

  ipa_point_prep<<<(LSEQ * NH + 255) / 256, 256, 0, stream>>>(
      q1, k1, q2p, k2p, v3p, rot, trans, gamma, Aex, Bex, v3g);

  ipa_fused<<<LSEQ, 256, 0, stream>>>(z, Wz, Aex, Bex, v1, v3g, rot, trans,
                                      o1w, o2w, o3w);

  ipa_init_out<<<(LSEQ * DIN + 255) / 256, 256, 0, stream>>>(b1, b2, b3, out);
  gemm(o1w, W1, out, LSEQ, DIN, NH * DZ,      1);
  gemm(o2w, W2, out, LSEQ, DIN, NH * CD,      1);
  gemm(o3w, W3, out, LSEQ, DIN, NH * NPP * 3, 1);
}